// SimpleGATLayer_15556371546815
// MI455X (gfx1250) — compile-verified
//
#include <hip/hip_runtime.h>
#include <hip/hip_bf16.h>

typedef __attribute__((ext_vector_type(2))) float v2f;
typedef __attribute__((ext_vector_type(8))) float v8f;
typedef __attribute__((ext_vector_type(2))) int   v2i;
typedef __attribute__((ext_vector_type(4))) int   v4i;

constexpr int NB   = 32;    // batch
constexpr int NK   = 1024;  // nodes
constexpr int NFIN = 256;
constexpr int NFO  = 128;   // F_OUT
constexpr float NEG_INF = -9.0e15f;
constexpr float ALPHA = 0.2f;

#define AS1 __attribute__((address_space(1)))
#define AS3 __attribute__((address_space(3)))

#if defined(__has_builtin)
#if __has_builtin(__builtin_amdgcn_global_load_async_to_lds_b128) && \
    __has_builtin(__builtin_amdgcn_s_wait_asynccnt)
#define HAVE_ASYNC_LDS 1
#endif
#endif
#ifndef HAVE_ASYNC_LDS
#define HAVE_ASYNC_LDS 0
#endif

// ---------------------------------------------------------------------------
// Kernel 1: Wh = h @ W   (per-batch [1024,256] x [256,128])
// One wave per 16x16 output tile; fp32 WMMA 16x16x4, 64 steps over K=256.
// A layout: lane%16 = M row; VGPR0/1 hold K = kk+2g, kk+2g+1  (g = lane/16)
// B layout: lane%16 = N col; VGPR0/1 hold K = kk+2g, kk+2g+1
// C layout: lane%16 = N col; VGPR r holds M = r + 8g
// ---------------------------------------------------------------------------
__global__ __launch_bounds__(128) void gat_wh_kernel(const float* __restrict__ h,
                                                     const float* __restrict__ W,
                                                     float* __restrict__ Wh) {
    const int lane = threadIdx.x & 31;
    const int wid  = blockIdx.x * 4 + (threadIdx.x >> 5);   // 32*64*8 waves
    const int b  = wid >> 9;         // / (64*8)
    const int mt = (wid >> 3) & 63;  // row tile
    const int nt = wid & 7;          // col tile
    const int r  = lane & 15;
    const int g  = lane >> 4;

    const float* hrow = h + (size_t)(b * NK + mt * 16 + r) * NFIN;
    const float* wcol = W + nt * 16 + r;

    v8f acc = {};
    for (int kk = 0; kk < NFIN; kk += 4) {
        // h is streamed exactly once -> non-temporal
        v2f a = __builtin_nontemporal_load((const v2f*)(hrow + kk + 2 * g));
        v2f bb;
        bb.x = wcol[(size_t)(kk + 2 * g)     * NFO];
        bb.y = wcol[(size_t)(kk + 2 * g + 1) * NFO];
        acc = __builtin_amdgcn_wmma_f32_16x16x4_f32(false, a, false, bb,
                                                    (short)0, acc, false, false);
    }

    float* outp = Wh + (size_t)(b * NK + mt * 16) * NFO + nt * 16 + r;
#pragma unroll
    for (int vr = 0; vr < 8; ++vr)
        outp[(size_t)(vr + 8 * g) * NFO] = acc[vr];   // Wh stays cache-resident
}

// ---------------------------------------------------------------------------
// Kernel 2: src = Wh @ a1, dst = Wh @ a2   (one wave per node row)
// ---------------------------------------------------------------------------
__global__ __launch_bounds__(128) void gat_srcdst_kernel(const float* __restrict__ Wh,
                                                         const float* __restrict__ a1,
                                                         const float* __restrict__ a2,
                                                         float* __restrict__ src,
                                                         float* __restrict__ dst) {
    const int lane = threadIdx.x & 31;
    const int row  = blockIdx.x * 4 + (threadIdx.x >> 5);   // NB*NK rows
    const float* wr = Wh + (size_t)row * NFO;
    float s1 = 0.f, s2 = 0.f;
#pragma unroll
    for (int q = 0; q < NFO; q += 32) {
        float v = wr[q + lane];
        s1 += v * a1[q + lane];
        s2 += v * a2[q + lane];
    }
#pragma unroll
    for (int off = 16; off > 0; off >>= 1) {
        s1 += __shfl_xor(s1, off, 32);
        s2 += __shfl_xor(s2, off, 32);
    }
    if (lane == 0) { src[row] = s1; dst[row] = s2; }
}

// ---------------------------------------------------------------------------
// Async (or fallback) staging of one 16x128 Wh tile (8 KB) into LDS.
// 128 threads x 4 x b128 = 8 KB per call.
// ---------------------------------------------------------------------------
__device__ __forceinline__ void stage_wh_tile(const float* __restrict__ gsrc,
                                              float* lbuf, int tid) {
    const float4* sp = (const float4*)gsrc;
    float4* dp = (float4*)lbuf;
#if HAVE_ASYNC_LDS
#pragma unroll
    for (int q = 0; q < 4; ++q) {
        __builtin_amdgcn_global_load_async_to_lds_b128(
            (AS1 v4i*)(uintptr_t)(sp + tid + 128 * q),
            (AS3 v4i*)(uintptr_t)(dp + tid + 128 * q),
            /*offset=*/0, /*cpol=*/0);
    }
#else
#pragma unroll
    for (int q = 0; q < 4; ++q) dp[tid + 128 * q] = sp[tid + 128 * q];
#endif
}

__device__ __forceinline__ void stage_wait() {
#if HAVE_ASYNC_LDS
    __builtin_amdgcn_s_wait_asynccnt(0);
#endif
    __syncthreads();
}

// ---------------------------------------------------------------------------
// Kernel 3: fused mask + online-softmax + att@Wh + ELU.
// 4 waves / workgroup, each wave owns one 16-row tile and all 128 out cols.
// Wh j-tiles double-buffered in LDS via async global->LDS copies; adj (134 MB,
// the bandwidth-dominant stream) read exactly once with non-temporal loads.
// ---------------------------------------------------------------------------
__global__ __launch_bounds__(128) void gat_softmax_aggr_kernel(
        const int* __restrict__ adj, const float* __restrict__ Wh,
        const float* __restrict__ src, const float* __restrict__ dst,
        float* __restrict__ out) {
    __shared__ __align__(16) float lds_wh[2][16 * NFO];  // 2 x 8 KB
    __shared__ __align__(16) float lds_dst[NK];          // 4 KB

    const int tid  = threadIdx.x;
    const int lane = tid & 31;
    const int w    = tid >> 5;
    const int b    = blockIdx.x >> 4;       // 16 row-groups per batch
    const int rg   = blockIdx.x & 15;
    const int tile_i = rg * 4 + w;
    const int r = lane & 15;
    const int g = lane >> 4;
    const int row = tile_i * 16 + r;

    // stage dst[b, :] into LDS; stage first Wh tile; one barrier covers both
    for (int q = tid; q < NK; q += 128) lds_dst[q] = dst[b * NK + q];
    stage_wh_tile(Wh + (size_t)(b * NK) * NFO, lds_wh[0], tid);
    stage_wait();

    const float src_i = src[b * NK + row];
    const int* adj_row = adj + (size_t)(b * NK + row) * NK;

    v8f acc[8];
#pragma unroll
    for (int nt = 0; nt < 8; ++nt) acc[nt] = {};
    float m_old = -__builtin_inff();
    float l = 0.f;

    for (int jt = 0; jt < NK / 16; ++jt) {
        const int cur = jt & 1;
        // prefetch next tile into the other buffer (uniform branch)
        if (jt + 1 < NK / 16)
            stage_wh_tile(Wh + (size_t)(b * NK + (jt + 1) * 16) * NFO,
                          lds_wh[cur ^ 1], tid);
        const float* wtile = lds_wh[cur];

        // scores for this lane: row = tile row r, j = jt*16 + 4c + 2g + {0,1}
        float e[4][2];
#pragma unroll
        for (int c = 0; c < 4; ++c) {
            const int joff = 4 * c + 2 * g;
            const float2 dv = *(const float2*)(lds_dst + jt * 16 + joff);
            const v2i av = __builtin_nontemporal_load(
                (const v2i*)(adj_row + jt * 16 + joff));
            float e0 = src_i + dv.x;
            float e1 = src_i + dv.y;
            e0 = e0 > 0.f ? e0 : ALPHA * e0;
            e1 = e1 > 0.f ? e1 : ALPHA * e1;
            e[c][0] = (av[0] > 0) ? e0 : NEG_INF;
            e[c][1] = (av[1] > 0) ? e1 : NEG_INF;
        }

        // online softmax: tile row-max (combine the two half-wave j-sets)
        float mt_ = e[0][0];
#pragma unroll
        for (int c = 0; c < 4; ++c) {
            mt_ = fmaxf(mt_, e[c][0]);
            mt_ = fmaxf(mt_, e[c][1]);
        }
        mt_ = fmaxf(mt_, __shfl_xor(mt_, 16, 32));
        const float m_new = fmaxf(m_old, mt_);
        const float rf = expf(m_old - m_new);   // first iter: expf(-inf)=0

        float p[4][2];
        float s = 0.f;
#pragma unroll
        for (int c = 0; c < 4; ++c) {
            p[c][0] = expf(e[c][0] - m_new);
            p[c][1] = expf(e[c][1] - m_new);
            s += p[c][0] + p[c][1];
        }
        s += __shfl_xor(s, 16, 32);
        l = l * rf + s;
        m_old = m_new;

        // rescale accumulators: acc[nt][vr] belongs to row (vr + 8g)
        float rfr[8];
#pragma unroll
        for (int vr = 0; vr < 8; ++vr) rfr[vr] = __shfl(rf, vr + 8 * g, 32);
#pragma unroll
        for (int nt = 0; nt < 8; ++nt)
#pragma unroll
            for (int vr = 0; vr < 8; ++vr) acc[nt][vr] *= rfr[vr];

        // acc += P(16x16) x WhTile(16x128): 4 K-chunks x 8 N-tiles of WMMA
#pragma unroll
        for (int c = 0; c < 4; ++c) {
            v2f a;
            a.x = p[c][0];
            a.y = p[c][1];
            const int krow = 4 * c + 2 * g;
#pragma unroll
            for (int nt = 0; nt < 8; ++nt) {
                v2f bb;
                bb.x = wtile[krow * NFO + nt * 16 + r];
                bb.y = wtile[(krow + 1) * NFO + nt * 16 + r];
                acc[nt] = __builtin_amdgcn_wmma_f32_16x16x4_f32(
                    false, a, false, bb, (short)0, acc[nt], false, false);
            }
        }

        // next iteration reads the prefetched buffer: drain async + barrier
        stage_wait();
    }

    // normalize (1/l per row), ELU, store (output streamed once -> NT stores)
    const float invl = 1.0f / l;    // lane holds row r's sum
    float inv[8];
#pragma unroll
    for (int vr = 0; vr < 8; ++vr) inv[vr] = __shfl(invl, vr + 8 * g, 32);

    float* outp = out + (size_t)(b * NK + tile_i * 16) * NFO + r;
#pragma unroll
    for (int nt = 0; nt < 8; ++nt)
#pragma unroll
        for (int vr = 0; vr < 8; ++vr) {
            float v = acc[nt][vr] * inv[vr];
            v = v > 0.f ? v : (expf(v) - 1.0f);
            __builtin_nontemporal_store(v, outp + (size_t)(vr + 8 * g) * NFO + nt * 16);
        }
}

// ---------------------------------------------------------------------------
extern "C" void kernel_launch(void* const* d_in, const int* in_sizes, int n_in,
                              void* d_out, int out_size, void* d_ws, size_t ws_size,
                              hipStream_t stream) {
    const float* h   = (const float*)d_in[0];
    const int*   adj = (const int*)d_in[1];
    const float* W   = (const float*)d_in[2];
    const float* a1  = (const float*)d_in[3];
    const float* a2  = (const float*)d_in[4];
    float* out = (float*)d_out;

    float* Wh  = (float*)d_ws;                        // NB*NK*NFO floats (16.8 MB)
    float* src = Wh + (size_t)NB * NK * NFO;          // NB*NK
    float* dst = src + (size_t)NB * NK;               // NB*NK

    // Kernel 1: 32*64*8 = 16384 waves, 4 waves/block
    gat_wh_kernel<<<4096, 128, 0, stream>>>(h, W, Wh);
    // Kernel 2: 32768 rows, 4 rows/block
    gat_srcdst_kernel<<<8192, 128, 0, stream>>>(Wh, a1, a2, src, dst);
    // Kernel 3: 32 batches * 16 row-groups, 4 row-tiles/block
    gat_softmax_aggr_kernel<<<512, 128, 0, stream>>>(adj, Wh, src, dst, out);
}